// NStepQValueLoss_16303695855652
// MI455X (gfx1250) — compile-verified
//
#include <hip/hip_runtime.h>
#include <cstddef>

// N-step Q-value loss, B=4096, T=512, Q=4, fp32.
// Memory-bound streaming kernel: ~160MB traffic -> ~6.9us floor at 23.3 TB/s.
// CDNA5 path: global_load_async_to_lds_b128 staging of halo-shared rows + s_wait_asynccnt.

typedef float v4f __attribute__((ext_vector_type(4)));

#define GAMMA_F 0.997f
#define EPS_F   1e-3f
#define TT      512
#define QQ      4
// log2(0.997) in fp32
#define LOG2_GAMMA (-0.0043347017f)

__device__ __forceinline__ float sgnf(float x) {
    // true three-way sign: jnp.sign(0) == 0 (matters for inv_rescale at x==0)
    return (x > 0.f) ? 1.f : ((x < 0.f) ? -1.f : 0.f);
}

__device__ __forceinline__ float rescale1(float x) {
    return sgnf(x) * (__builtin_sqrtf(__builtin_fabsf(x) + 1.f) - 1.f) + EPS_F * x;
}

__device__ __forceinline__ float inv_rescale1(float x) {
    float sa = __builtin_fmaf(4.f * EPS_F, __builtin_fabsf(x) + (1.f + EPS_F), 1.f);
    float r  = (__builtin_sqrtf(sa) - 1.f) * (1.f / (2.f * EPS_F));
    return sgnf(x) * __builtin_fmaf(r, r, -1.f);
}

// low-32-bit LDS byte offset of a __shared__ object (addrspace(3) pointers are 32-bit)
__device__ __forceinline__ unsigned lds_addr32(const void* p) {
    return (unsigned)(size_t)(__attribute__((address_space(3))) const void*)p;
}

__global__ __launch_bounds__(TT) void nstep_qloss_kernel(
    const float* __restrict__ cur_q,    // (B,T,Q)
    const float* __restrict__ next_q,   // (B,T,Q)
    const float* __restrict__ log_p,    // (B,T)
    const float* __restrict__ reward,   // (B,T,Q)
    const float* __restrict__ is_done,  // (B,T)
    const float* __restrict__ mask,     // (B,T)
    float* __restrict__ out)            // (B,T,Q)
{
    __shared__ __align__(16) v4f   s_r[TT];   // reward row (later: masked reward)
    __shared__ __align__(16) float s_m[TT];   // mask row
    __shared__ __align__(16) float s_d[TT];   // is_done row
    __shared__ __align__(16) float s_lp[TT];  // log_p row
    __shared__ __align__(16) float s_s[TT];   // m*(1-d)*log_p

    const int    t    = threadIdx.x;
    const int    b    = blockIdx.x;
    const size_t rowq = (size_t)b * TT * QQ;  // float index of row start, Q-major arrays
    const size_t row  = (size_t)b * TT;       // float index of row start, scalar arrays

    // ---- async stage: global -> LDS (CDNA5 async data mover, ASYNCcnt) ----
    {
        unsigned voff = (unsigned)t * 16u;  // byte offset, one b128 per lane
        unsigned l_r  = lds_addr32(&s_r[t]);
        unsigned long long base_r = (unsigned long long)(size_t)(reward + rowq);
        asm volatile("global_load_async_to_lds_b128 %0, %1, %2"
                     :: "v"(l_r), "v"(voff), "s"(base_r) : "memory");
        if (t < TT / 4) {  // 512 floats per scalar row = 128 b128 transfers
            unsigned l_m = lds_addr32(&s_m[t * 4]);
            unsigned long long base_m = (unsigned long long)(size_t)(mask + row);
            asm volatile("global_load_async_to_lds_b128 %0, %1, %2"
                         :: "v"(l_m), "v"(voff), "s"(base_m) : "memory");
            unsigned l_d = lds_addr32(&s_d[t * 4]);
            unsigned long long base_d = (unsigned long long)(size_t)(is_done + row);
            asm volatile("global_load_async_to_lds_b128 %0, %1, %2"
                         :: "v"(l_d), "v"(voff), "s"(base_d) : "memory");
            unsigned l_l = lds_addr32(&s_lp[t * 4]);
            unsigned long long base_l = (unsigned long long)(size_t)(log_p + row);
            asm volatile("global_load_async_to_lds_b128 %0, %1, %2"
                         :: "v"(l_l), "v"(voff), "s"(base_l) : "memory");
        }
    }
    asm volatile("s_wait_asynccnt 0x0" ::: "memory");  // own wave's transfers done
    __syncthreads();                                   // everyone's transfers done

    // ---- phase 1: mask reward in place, precompute s = m*(1-d)*log_p ----
    const float m = s_m[t];
    const float d = s_d[t];
    {
        v4f r = s_r[t];
        s_r[t] = m * r;
        s_s[t] = m * (1.f - d) * s_lp[t];
    }
    __syncthreads();

    // ---- phase 2: 5-tap discounted window sums (zero-padded past T) ----
    const float gw[5] = {1.f, GAMMA_F, GAMMA_F * GAMMA_F,
                         GAMMA_F * GAMMA_F * GAMMA_F,
                         GAMMA_F * GAMMA_F * GAMMA_F * GAMMA_F};
    v4f  rs = {0.f, 0.f, 0.f, 0.f};
    float ls = 0.f;
#pragma unroll
    for (int n = 0; n < 5; ++n) {
        int   j = t + n;
        float w = (j < TT) ? gw[n] : 0.f;
        int  jj = (j < TT) ? j : (TT - 1);
        v4f   r = s_r[jj];
        rs += w * r;
        ls  = __builtin_fmaf(w, s_s[jj], ls);
    }
    const float lsg = ls * GAMMA_F;  // log_p window weight is gamma^(n+1)

    // ---- bootstrap term at idx = min(T-1, t+4), weight gamma^idx (absolute) ----
    const int   idx = (t + 4 < TT) ? (t + 4) : (TT - 1);
    const float mi  = s_m[idx];
    const float di  = s_d[idx];
    const v4f   nq  = __builtin_nontemporal_load((const v4f*)(next_q + rowq) + idx);
    const float gpi = __builtin_exp2f((float)idx * LOG2_GAMMA);
    const float f   = gpi * mi * (1.f - di);

    // per-q constants: qw = (1, 0.5, 0, 2); c = inv_num_q / qw = (1/3, 2/3, 0, 1/6)
    float px = rs.x + f * inv_rescale1(nq.x) + (1.f / 3.f) * lsg;
    float py = rs.y + f * inv_rescale1(nq.y) + (2.f / 3.f) * lsg;
    float pw = rs.w + f * inv_rescale1(nq.w) + (1.f / 6.f) * lsg;

    const v4f cq = __builtin_nontemporal_load((const v4f*)(cur_q + rowq) + t);
    const float hm = 0.5f * m;

    float ex = cq.x - rescale1(px);
    float ey = cq.y - rescale1(py);
    float ew = cq.w - rescale1(pw);

    v4f L;
    L.x = (hm * 1.0f) * ex * ex;
    L.y = (hm * 0.5f) * ey * ey;
    L.z = 0.f;                      // qw==0 -> loss component exactly 0
    L.w = (hm * 2.0f) * ew * ew;

    __builtin_nontemporal_store(L, (v4f*)out + row + t);
}

extern "C" void kernel_launch(void* const* d_in, const int* in_sizes, int n_in,
                              void* d_out, int out_size, void* d_ws, size_t ws_size,
                              hipStream_t stream) {
    (void)n_in; (void)out_size; (void)d_ws; (void)ws_size;
    const float* cur  = (const float*)d_in[0];  // current_q_value (B,T,Q)
    const float* nxt  = (const float*)d_in[1];  // next_q_value    (B,T,Q)
    const float* lp   = (const float*)d_in[2];  // log_p           (B,T)
    const float* rew  = (const float*)d_in[3];  // reward          (B,T,Q)
    const float* done = (const float*)d_in[4];  // is_done         (B,T)
    const float* msk  = (const float*)d_in[5];  // mask            (B,T)
    float* out = (float*)d_out;

    const int B = in_sizes[2] / TT;  // log_p is (B,T)
    nstep_qloss_kernel<<<B, TT, 0, stream>>>(cur, nxt, lp, rew, done, msk, out);
}